// LTCCell_75411035783326
// MI455X (gfx1250) — compile-verified
//
#include <hip/hip_runtime.h>
#include <stdint.h>

// ---------------- problem constants (match reference) ----------------
#define B_TOT   4096
#define S_DIM   64
#define N_DIM   128
#define M_DIM   32
#define UNFOLDS 6
#define EPS_F   1e-8f
#define LOG2E   1.44269504088896340736f

#define BT      16              // batch tile per workgroup
#define THREADS 256             // 8 waves (wave32)
#define NN      (N_DIM * N_DIM) // 16384

// LDS layout (floats):
//   [0      .. 65536)  fused params, float4-interleaved {wm, wrev, sigL, smuL}  (256 KB)
//   [65536  .. 67584)  cur tile  [BT][N]                                        (  8 KB)
//   [67584  .. 68608)  xt  tile  [BT][S]                                        (  4 KB)
#define LDS_FLOATS 68608
#define LDS_BYTES  (LDS_FLOATS * 4)   // 274432 < 320 KB WGP LDS

// ---------------------------------------------------------------------
// Kernel 1: fold recurrent params once into workspace as interleaved float4.
//   sigmoid(sigma*(c-mu)) = 1 / (1 + exp2(smuL - sigL*c))
// ---------------------------------------------------------------------
__global__ __launch_bounds__(256) void ltc_fuse_params(
    const float* __restrict__ w,     const float* __restrict__ sigma,
    const float* __restrict__ mu,    const float* __restrict__ rev,
    const float* __restrict__ mask,  float4* __restrict__ ws4)
{
    int idx = blockIdx.x * 256 + threadIdx.x;   // grid = 64*256 == NN exactly
    float wm = w[idx] * mask[idx];
    float sg = sigma[idx];
    ws4[idx] = make_float4(wm, wm * rev[idx], sg * LOG2E, sg * mu[idx] * LOG2E);
}

// ---------------------------------------------------------------------
// Kernel 2: full LTC cell for a 16-batch tile.
// ---------------------------------------------------------------------
template <bool USE_WS>
__global__ __launch_bounds__(THREADS) void ltc_cell(
    const float* __restrict__ x,      const float* __restrict__ state,
    const float* __restrict__ gleak,  const float* __restrict__ erev,
    const float* __restrict__ mcap,
    const float* __restrict__ s_w,    const float* __restrict__ s_sigma,
    const float* __restrict__ s_mu,   const float* __restrict__ s_rev,
    const float* __restrict__ s_mask,
    const float* __restrict__ iw,     const float* __restrict__ ib,
    const float* __restrict__ ow,     const float* __restrict__ ob,
    const int*   __restrict__ tc,
    const float* __restrict__ ws,                      // fused params (float4-packed)
    const float* __restrict__ w,      const float* __restrict__ sigma,
    const float* __restrict__ mu,     const float* __restrict__ rev,
    const float* __restrict__ mask,                    // raw params (fallback path)
    float* __restrict__ out_motor,    // [B, M]
    float* __restrict__ out_state)    // [B, N]
{
    extern __shared__ float smem[];
    float4* p4    = (float4*)smem;          // [N][N] of {wm, wrev, sigL, smuL}
    float*  cur_t = smem + 4 * NN;          // [BT][N]
    float*  xt_t  = smem + 4 * NN + BT * N_DIM; // [BT][S]

    const int tid  = threadIdx.x;
    const int j    = tid & (N_DIM - 1);     // neuron index owned by this thread
    const int bsub = tid >> 7;              // 0..1 -> this thread handles 8 batches
    const int b0   = blockIdx.x * BT;

    // ---- stage fused params into LDS ----
    if (USE_WS) {
        // CDNA5 async DMA: global -> LDS, 16B per lane per op, tracked by ASYNCcnt.
        // Overlapped with the sensory pass below; waited on before the unfolds.
        const uint32_t ldsbase = (uint32_t)(uintptr_t)(void*)p4; // low 32b of generic LDS ptr = LDS offset
        const uint64_t gbase   = (uint64_t)(uintptr_t)ws;
        #pragma unroll 4
        for (int r = 0; r < (4 * NN * 4) / (THREADS * 16); ++r) {   // 64 rounds * 4KB
            uint32_t off   = (uint32_t)(r * THREADS + tid) * 16u;
            uint32_t laddr = ldsbase + off;
            asm volatile("global_load_async_to_lds_b128 %0, %1, %2 offset:0"
                         :: "v"(laddr), "v"(off), "s"(gbase) : "memory");
        }
    } else {
        // Fallback: fold params straight into LDS (no workspace needed).
        for (int t = tid; t < NN; t += THREADS) {
            float wm = w[t] * mask[t];
            float sg = sigma[t];
            p4[t] = make_float4(wm, wm * rev[t], sg * LOG2E, sg * mu[t] * LOG2E);
        }
    }

    // ---- stage xt (affine input map) and cur (state) tiles ----
    for (int t = tid; t < BT * S_DIM; t += THREADS) {
        int bb = t >> 6, ss = t & (S_DIM - 1);
        xt_t[t] = x[(b0 + bb) * S_DIM + ss] * iw[ss] + ib[ss];
    }
    for (int t = tid; t < BT * N_DIM; t += THREADS) {
        int bb = t >> 7, jj = t & (N_DIM - 1);
        cur_t[t] = state[(b0 + bb) * N_DIM + jj];
    }
    __syncthreads();   // xt/cur tiles visible (params may still be in flight)

    // ---- per-neuron constants ----
    const float tcf    = (float)(*tc);
    const float cap    = mcap[j] * ((float)UNFOLDS / tcf);  // membrane_cap / (tc/UNFOLDS)
    const float gl     = gleak[j];
    const float glerev = gl * erev[j];

    // ---- sensory synapse pass (once; overlaps the async param DMA) ----
    float snum[8], sden[8];
    #pragma unroll
    for (int k = 0; k < 8; ++k) { snum[k] = 0.f; sden[k] = 0.f; }

    for (int i = 0; i < S_DIM; ++i) {
        const int base = i * N_DIM + j;
        if (i + 8 < S_DIM) {                 // gfx1250 global_prefetch_b8
            __builtin_prefetch(s_w     + base + 8 * N_DIM, 0, 0);
            __builtin_prefetch(s_sigma + base + 8 * N_DIM, 0, 0);
        }
        float swm = s_w[base] * s_mask[base];
        float sl  = s_sigma[base] * LOG2E;
        float a2  = sl * s_mu[base];
        float swr = swm * s_rev[base];
        #pragma unroll
        for (int k = 0; k < 8; ++k) {
            float xv = xt_t[(bsub * 8 + k) * S_DIM + i];
            float e  = __builtin_amdgcn_exp2f(fmaf(-sl, xv, a2)); // exp(-sig*(x-mu))
            float r  = __builtin_amdgcn_rcpf(1.0f + e);           // sigmoid
            sden[k]  = fmaf(swm, r, sden[k]);
            snum[k]  = fmaf(swr, r, snum[k]);
        }
    }

    if (USE_WS) {
        asm volatile("s_wait_asynccnt 0" ::: "memory");  // our async LDS writes landed
    }
    __syncthreads();                                     // everyone's chunk landed

    // ---- 6 ODE unfolds, all operands in LDS ----
    for (int u = 0; u < UNFOLDS; ++u) {
        float num[8], den[8];
        #pragma unroll
        for (int k = 0; k < 8; ++k) { num[k] = 0.f; den[k] = 0.f; }

        for (int i0 = 0; i0 < N_DIM; i0 += 4) {
            float4 pq0 = p4[(i0 + 0) * N_DIM + j];   // one ds_load_b128: all 4 params
            float4 pq1 = p4[(i0 + 1) * N_DIM + j];
            float4 pq2 = p4[(i0 + 2) * N_DIM + j];
            float4 pq3 = p4[(i0 + 3) * N_DIM + j];
            #pragma unroll
            for (int k = 0; k < 8; ++k) {
                float4 c4 = *(const float4*)&cur_t[(bsub * 8 + k) * N_DIM + i0]; // broadcast b128
                float e, r;
                e = __builtin_amdgcn_exp2f(fmaf(-pq0.z, c4.x, pq0.w));
                r = __builtin_amdgcn_rcpf(1.0f + e);
                den[k] = fmaf(pq0.x, r, den[k]);  num[k] = fmaf(pq0.y, r, num[k]);
                e = __builtin_amdgcn_exp2f(fmaf(-pq1.z, c4.y, pq1.w));
                r = __builtin_amdgcn_rcpf(1.0f + e);
                den[k] = fmaf(pq1.x, r, den[k]);  num[k] = fmaf(pq1.y, r, num[k]);
                e = __builtin_amdgcn_exp2f(fmaf(-pq2.z, c4.z, pq2.w));
                r = __builtin_amdgcn_rcpf(1.0f + e);
                den[k] = fmaf(pq2.x, r, den[k]);  num[k] = fmaf(pq2.y, r, num[k]);
                e = __builtin_amdgcn_exp2f(fmaf(-pq3.z, c4.w, pq3.w));
                r = __builtin_amdgcn_rcpf(1.0f + e);
                den[k] = fmaf(pq3.x, r, den[k]);  num[k] = fmaf(pq3.y, r, num[k]);
            }
        }

        float newc[8];
        #pragma unroll
        for (int k = 0; k < 8; ++k) {
            float c  = cur_t[(bsub * 8 + k) * N_DIM + j];
            float dh = fmaf(cap, c, glerev) + num[k] + snum[k];
            float dt = cap + gl + den[k] + sden[k] + EPS_F;
            newc[k]  = dh * __builtin_amdgcn_rcpf(dt);
        }
        __syncthreads();                          // all reads of old cur done
        #pragma unroll
        for (int k = 0; k < 8; ++k)
            cur_t[(bsub * 8 + k) * N_DIM + j] = newc[k];
        __syncthreads();                          // new cur visible
    }

    // ---- outputs: motor map on first M neurons + final state ----
    float owj = (j < M_DIM) ? ow[j] : 0.f;
    float obj = (j < M_DIM) ? ob[j] : 0.f;
    #pragma unroll
    for (int k = 0; k < 8; ++k) {
        int   b = b0 + bsub * 8 + k;
        float c = cur_t[(bsub * 8 + k) * N_DIM + j];
        out_state[b * N_DIM + j] = c;
        if (j < M_DIM) out_motor[b * M_DIM + j] = fmaf(c, owj, obj);
    }
}

// ---------------------------------------------------------------------
extern "C" void kernel_launch(void* const* d_in, const int* in_sizes, int n_in,
                              void* d_out, int out_size, void* d_ws, size_t ws_size,
                              hipStream_t stream) {
    const float* x      = (const float*)d_in[0];
    const float* state  = (const float*)d_in[1];
    const float* gleak  = (const float*)d_in[2];
    const float* erev   = (const float*)d_in[3];
    const float* mcap   = (const float*)d_in[4];
    const float* w      = (const float*)d_in[5];
    const float* sigma  = (const float*)d_in[6];
    const float* mu     = (const float*)d_in[7];
    const float* srp    = (const float*)d_in[8];
    const float* sw     = (const float*)d_in[9];
    const float* ssig   = (const float*)d_in[10];
    const float* smu    = (const float*)d_in[11];
    const float* srev   = (const float*)d_in[12];
    const float* mask   = (const float*)d_in[13];
    const float* smask  = (const float*)d_in[14];
    const float* iw     = (const float*)d_in[15];
    const float* ib     = (const float*)d_in[16];
    const float* ow     = (const float*)d_in[17];
    const float* ob     = (const float*)d_in[18];
    const int*   tc     = (const int*)d_in[19];

    float* out_motor = (float*)d_out;
    float* out_state = out_motor + (size_t)B_TOT * M_DIM;

    const bool use_ws = (ws_size >= (size_t)(4 * NN * sizeof(float)));

    if (use_ws) {
        ltc_fuse_params<<<NN / 256, 256, 0, stream>>>(w, sigma, mu, srp, mask,
                                                      (float4*)d_ws);
        hipFuncSetAttribute((const void*)ltc_cell<true>,
                            hipFuncAttributeMaxDynamicSharedMemorySize, LDS_BYTES);
        ltc_cell<true><<<B_TOT / BT, THREADS, LDS_BYTES, stream>>>(
            x, state, gleak, erev, mcap, sw, ssig, smu, srev, smask,
            iw, ib, ow, ob, tc, (const float*)d_ws,
            w, sigma, mu, srp, mask, out_motor, out_state);
    } else {
        hipFuncSetAttribute((const void*)ltc_cell<false>,
                            hipFuncAttributeMaxDynamicSharedMemorySize, LDS_BYTES);
        ltc_cell<false><<<B_TOT / BT, THREADS, LDS_BYTES, stream>>>(
            x, state, gleak, erev, mcap, sw, ssig, smu, srev, smask,
            iw, ib, ow, ob, tc, (const float*)d_ws,
            w, sigma, mu, srp, mask, out_motor, out_state);
    }
}